// SegmentSum_20272245637565
// MI455X (gfx1250) — compile-verified
//
#include <hip/hip_runtime.h>

#define D_FEAT 128
#define WAVES_PER_BLOCK 4
#define BLOCK_THREADS (WAVES_PER_BLOCK * 32)
#define TILE_ROWS 16
#define TILES_PER_WAVE 16
#define ROWS_PER_WAVE (TILE_ROWS * TILES_PER_WAVE)  // 256 rows per wave

#if defined(__HIP_DEVICE_COMPILE__) && \
    __has_builtin(__builtin_amdgcn_global_load_async_to_lds_b128) && \
    __has_builtin(__builtin_amdgcn_s_wait_asynccnt)
#define USE_ASYNC 1
#else
#define USE_ASYNC 0
#endif

#if USE_ASYNC
typedef int v4i __attribute__((vector_size(16)));
typedef __attribute__((address_space(1))) v4i* as1_v4i;
typedef __attribute__((address_space(3))) v4i* as3_v4i;
// gfx12+ CPol: TH in bits [2:0]; TH=1 -> non-temporal (x is streamed exactly
// once: keep the 192MB L2 for the atomically-updated output + index instead)
#define XLOAD_CPOL 1
#endif

__device__ __forceinline__ void flush_seg(float* __restrict__ out, unsigned seg,
                                          unsigned col, const float4& acc) {
    float* o = out + (size_t)(seg * (unsigned)D_FEAT + col);
    __hip_atomic_fetch_add(o + 0, acc.x, __ATOMIC_RELAXED, __HIP_MEMORY_SCOPE_AGENT);
    __hip_atomic_fetch_add(o + 1, acc.y, __ATOMIC_RELAXED, __HIP_MEMORY_SCOPE_AGENT);
    __hip_atomic_fetch_add(o + 2, acc.z, __ATOMIC_RELAXED, __HIP_MEMORY_SCOPE_AGENT);
    __hip_atomic_fetch_add(o + 3, acc.w, __ATOMIC_RELAXED, __HIP_MEMORY_SCOPE_AGENT);
}

__global__ __launch_bounds__(BLOCK_THREADS) void SegSum_kernel(
        const float* __restrict__ x,
        const int*   __restrict__ index,
        float*       __restrict__ out,
        unsigned n_rows)
{
#if USE_ASYNC
    __shared__ float lds[WAVES_PER_BLOCK][2][TILE_ROWS][D_FEAT];  // 64 KB
#endif
    const unsigned lane = threadIdx.x & 31u;
    const unsigned wid  = threadIdx.x >> 5;
    const unsigned gwave = blockIdx.x * WAVES_PER_BLOCK + wid;
    const unsigned row_start = gwave * ROWS_PER_WAVE;
    if (row_start >= n_rows) return;
    const unsigned row_end = (row_start + ROWS_PER_WAVE < n_rows)
                                 ? row_start + ROWS_PER_WAVE : n_rows;
    const unsigned span   = row_end - row_start;
    const unsigned ntiles = span >> 4;       // /TILE_ROWS
    const unsigned rem    = span & 15u;      // %TILE_ROWS

    float4 acc = make_float4(0.f, 0.f, 0.f, 0.f);
    int cur_seg = -1;
    const unsigned col = lane * 4u;  // this lane's 4 feature columns

#if USE_ASYNC
    // Issue one 16-row tile: 16 async b128 ops; each op moves one full row
    // (32 lanes x 16B = 512B) from HBM into this wave's LDS buffer.
    auto issue_tile = [&](unsigned t, unsigned stage) {
        const float* src = x + (size_t)((row_start + t * TILE_ROWS) * (unsigned)D_FEAT + col);
        #pragma unroll
        for (int r = 0; r < TILE_ROWS; ++r) {
            __builtin_amdgcn_global_load_async_to_lds_b128(
                (as1_v4i)(src + r * D_FEAT),
                (as3_v4i)&lds[wid][stage][r][col],
                /*offset=*/0, /*cpol=*/XLOAD_CPOL);
        }
    };
    if (ntiles > 0) issue_tile(0, 0);
#endif

    for (unsigned t = 0; t < ntiles; ++t) {
        const unsigned base = row_start + t * TILE_ROWS;
        // one coalesced 64B index fetch for the tile (lanes 16..31 duplicate 0..15)
        const int idxv = index[base + (lane & (TILE_ROWS - 1u))];

#if USE_ASYNC
        if (t + 1 < ntiles) {
            issue_tile(t + 1, (t + 1) & 1u);
            __builtin_amdgcn_s_wait_asynccnt(TILE_ROWS);  // tile t landed; t+1 in flight
        } else {
            __builtin_amdgcn_s_wait_asynccnt(0);
        }
        asm volatile("" ::: "memory");  // keep LDS reads after the wait
#endif
        #pragma unroll
        for (int r = 0; r < TILE_ROWS; ++r) {
            const int seg = __shfl(idxv, r, 32);  // wave-uniform
            if (seg != cur_seg) {
                if (cur_seg >= 0) flush_seg(out, (unsigned)cur_seg, col, acc);
                acc = make_float4(0.f, 0.f, 0.f, 0.f);
                cur_seg = seg;
            }
#if USE_ASYNC
            const float4 v = *(const float4*)&lds[wid][t & 1u][r][col];
#else
            const float4 v = *(const float4*)(x + (size_t)((base + r) * (unsigned)D_FEAT + col));
#endif
            acc.x += v.x; acc.y += v.y; acc.z += v.z; acc.w += v.w;
        }
    }

    // leftover rows (generality; N_ROWS is a multiple of 16 so normally rem==0)
    for (unsigned r = 0; r < rem; ++r) {
        const unsigned row = row_start + ntiles * TILE_ROWS + r;
        const int seg = index[row];  // broadcast load
        if (seg != cur_seg) {
            if (cur_seg >= 0) flush_seg(out, (unsigned)cur_seg, col, acc);
            acc = make_float4(0.f, 0.f, 0.f, 0.f);
            cur_seg = seg;
        }
        const float4 v = *(const float4*)(x + (size_t)(row * (unsigned)D_FEAT + col));
        acc.x += v.x; acc.y += v.y; acc.z += v.z; acc.w += v.w;
    }

    if (cur_seg >= 0) flush_seg(out, (unsigned)cur_seg, col, acc);
}

__global__ __launch_bounds__(256) void SegZero_kernel(float4* __restrict__ out, int n4) {
    int i = blockIdx.x * blockDim.x + threadIdx.x;
    if (i < n4) out[i] = make_float4(0.f, 0.f, 0.f, 0.f);
}

extern "C" void kernel_launch(void* const* d_in, const int* in_sizes, int n_in,
                              void* d_out, int out_size, void* d_ws, size_t ws_size,
                              hipStream_t stream) {
    const float* x     = (const float*)d_in[0];
    const int*   index = (const int*)d_in[1];
    float*       out   = (float*)d_out;
    const unsigned n_rows = (unsigned)in_sizes[1];   // 600000
    const int n4 = out_size / 4;                     // N_SEG*D_FEAT/4 float4s

    // 1) zero the output (empty segments must be 0; harness poisons d_out)
    SegZero_kernel<<<(n4 + 255) / 256, 256, 0, stream>>>((float4*)out, n4);

    // 2) segment sum
    const unsigned nwaves  = (n_rows + ROWS_PER_WAVE - 1) / ROWS_PER_WAVE;
    const unsigned nblocks = (nwaves + WAVES_PER_BLOCK - 1) / WAVES_PER_BLOCK;
    SegSum_kernel<<<nblocks, BLOCK_THREADS, 0, stream>>>(x, index, out, n_rows);
}